// SMAL_60739427500067
// MI455X (gfx1250) — compile-verified
//
#include <hip/hip_runtime.h>

#define NBATCH 1024
#define NVERT  3889
#define V3     11667       // NVERT*3
#define NJOINT 33
#define NROT   35
#define NBETA  41

typedef __attribute__((ext_vector_type(2))) float v2f;
typedef __attribute__((ext_vector_type(8))) float v8f;

__device__ __forceinline__ v8f wmma_f32x4(v2f a, v2f b, v8f c) {
  // D = A(16x4 f32) * B(4x16 f32) + C(16x16 f32)
  return __builtin_amdgcn_wmma_f32_16x16x4_f32(false, a, false, b, (short)0, c,
                                               false, false);
}

// ---------------------------------------------------------------------------
// K1: v_shaped = v_template + beta @ shapedirs     (1024x41)@(41x11667)
// One wave per 16x16 tile. Row tiles: 64, col tiles: 730 (tail 3 cols).
// Loads are unconditional with clamped addresses: garbage in a clamped B
// column only pollutes D elements in that same lane, which are never stored.
// ---------------------------------------------------------------------------
__global__ __launch_bounds__(256) void k1_shape(
    const float* __restrict__ beta, const float* __restrict__ shapedirs,
    const float* __restrict__ v_template, float* __restrict__ v_shaped) {
  const int lane = threadIdx.x & 31;
  const int wave = threadIdx.x >> 5;
  const int tile = blockIdx.x * 8 + wave;   // 46720 tiles total
  const int rt = tile / 730;
  const int ct = tile % 730;
  const int hl = lane & 15, hi = lane >> 4;
  const int row0 = rt * 16;
  const int col = ct * 16 + hl;
  const int colc = (col < V3) ? col : (V3 - 1);   // clamp, never branch
  const float* arow = beta + (row0 + hl) * NBETA;
  v8f c = {};
#pragma unroll
  for (int k0 = 0; k0 < 40; k0 += 4) {      // j0+1 <= 39 < 41: no guards
    const int j0 = k0 + 2 * hi;
    v2f a, b;
    a.x = arow[j0];
    a.y = arow[j0 + 1];
    b.x = shapedirs[(long)j0 * V3 + colc];
    b.y = shapedirs[(long)(j0 + 1) * V3 + colc];
    c = wmma_f32x4(a, b, c);
  }
  { // tail k0=40: only hi==0 lane-half carries k=40; k=41..43 are zero pad
    const float ta = arow[40];
    const float tb = shapedirs[(long)40 * V3 + colc];
    v2f a, b;
    a.x = hi ? 0.f : ta;  a.y = 0.f;
    b.x = hi ? 0.f : tb;  b.y = 0.f;
    c = wmma_f32x4(a, b, c);
  }
  if (col < V3) {
    const float vt = v_template[col];
#pragma unroll
    for (int r = 0; r < 8; ++r) {
      const int row = row0 + r + 8 * hi;     // C/D: VGPR r -> rows r, r+8
      v_shaped[(long)row * V3 + col] = c[r] + vt;
    }
  }
}

// ---------------------------------------------------------------------------
// K2 / K5: dst[n,j,a] = sum_v src[n,v,a]*Jreg[v,j] as a WMMA GEMM:
// rows = (n*3+a) -> 3072 (192 tiles), cols = j -> 33 (3 tiles, pad 48),
// K = v -> 3889 (972 full steps + 1 tail step).
// ---------------------------------------------------------------------------
__global__ __launch_bounds__(256) void k_jgemm(
    const float* __restrict__ src, const float* __restrict__ Jreg,
    float* __restrict__ dst, int dst_stride) {
  const int lane = threadIdx.x & 31;
  const int wave = threadIdx.x >> 5;
  const int tile = blockIdx.x * 8 + wave;   // 576 tiles total
  const int rt = tile / 3;
  const int ct = tile % 3;
  const int hl = lane & 15, hi = lane >> 4;
  const int row = rt * 16 + hl;             // n*3+a, < 3072 always
  const int col = ct * 16 + hl;             // j, pad to 48
  const int colc = (col < NJOINT) ? col : (NJOINT - 1);
  const float* arow = src + (long)(row / 3) * V3 + (row % 3);  // element v: arow[3v]
  v8f c = {};
  for (int k0 = 0; k0 < 3888; k0 += 4) {    // v0+1 <= 3887 < 3889: no guards
    const int v0 = k0 + 2 * hi;
    v2f a, b;
    a.x = arow[3 * v0];
    a.y = arow[3 * v0 + 3];
    b.x = Jreg[v0 * NJOINT + colc];
    b.y = Jreg[(v0 + 1) * NJOINT + colc];
    c = wmma_f32x4(a, b, c);
  }
  { // tail k0=3888: only hi==0 carries v=3888
    const float ta = arow[3 * 3888];
    const float tb = Jreg[3888 * NJOINT + colc];
    v2f a, b;
    a.x = hi ? 0.f : ta;  a.y = 0.f;
    b.x = hi ? 0.f : tb;  b.y = 0.f;
    c = wmma_f32x4(a, b, c);
  }
  if (col < NJOINT) {
#pragma unroll
    for (int r = 0; r < 8; ++r) {
      const int rowg = rt * 16 + r + 8 * hi;   // n*3+a
      dst[(long)(rowg / 3) * dst_stride + col * 3 + (rowg % 3)] = c[r];
    }
  }
}

// K5b: joints[n, 33+e, :] = verts[n, EXTRA[e], :]
__global__ __launch_bounds__(256) void k_extras(
    const float* __restrict__ verts, float* __restrict__ joints) {
  const int idx = blockIdx.x * 256 + threadIdx.x;   // 1024*18
  if (idx >= NBATCH * 18) return;
  const int extra_ids[6] = {1863, 26, 2124, 150, 3055, 1097};
  const int n = idx / 18, r = idx % 18;
  const int e = r / 3, a = r % 3;
  joints[(long)n * 117 + (NJOINT + e) * 3 + a] =
      verts[(long)n * V3 + extra_ids[e] * 3 + a];
}

// ---------------------------------------------------------------------------
// K3: Rodrigues + scale + kinematic chain -> A[n,33,16]. One thread per body.
// ---------------------------------------------------------------------------
__device__ __forceinline__ void rodrigues(const float* th, float R[3][3]) {
  const float tx = th[0], ty = th[1], tz = th[2];
  const float ax = tx + 1e-8f, ay = ty + 1e-8f, az = tz + 1e-8f;
  const float angle = sqrtf(ax * ax + ay * ay + az * az);
  const float inv = 1.f / angle;
  const float rx = tx * inv, ry = ty * inv, rz = tz * inv;
  const float cc = cosf(angle), ss = sinf(angle), oc = 1.f - cc;
  R[0][0] = cc + oc * rx * rx; R[0][1] = oc * rx * ry - ss * rz; R[0][2] = oc * rx * rz + ss * ry;
  R[1][0] = oc * ry * rx + ss * rz; R[1][1] = cc + oc * ry * ry; R[1][2] = oc * ry * rz - ss * rx;
  R[2][0] = oc * rz * rx - ss * ry; R[2][1] = oc * rz * ry + ss * rx; R[2][2] = cc + oc * rz * rz;
}

__device__ __forceinline__ float scal_of(int r, int c, const float* e) {
  const bool leg = (r >= 7 && r <= 14) || (r >= 17 && r <= 24);
  const bool tail = (r >= 25 && r <= 31);
  const bool ears = (r == 33 || r == 34);
  if (leg) return (c == 2) ? e[0] : e[1];
  if (tail) return (c == 0) ? e[2] : e[3];
  if (ears) return (c == 0) ? 1.f : ((c == 1) ? e[4] : e[5]);
  return 1.f;
}

__global__ __launch_bounds__(256) void k3_chain(
    const float* __restrict__ theta, const float* __restrict__ blog,
    const float* __restrict__ J_ws, float* __restrict__ A_ws) {
  const int n = blockIdx.x * blockDim.x + threadIdx.x;
  if (n >= NBATCH) return;
  float e[6];
#pragma unroll
  for (int s = 0; s < 6; ++s) e[s] = expf(blog[n * 6 + s]);
  const float* th = theta + (long)n * (NROT * 3);
  const float* Jn = J_ws + (long)n * (NJOINT * 3);
  float* An = A_ws + (long)n * (NJOINT * 16);

  float cur[3][4], R[3][3];
  rodrigues(th, R);
  float Jp[3] = {Jn[0], Jn[1], Jn[2]};
#pragma unroll
  for (int r = 0; r < 3; ++r) {
    cur[r][0] = R[r][0]; cur[r][1] = R[r][1]; cur[r][2] = R[r][2];
    cur[r][3] = Jp[r];
  }
  auto emit = [&](int i, const float* Jc) {
    float* o = An + i * 16;
#pragma unroll
    for (int r = 0; r < 3; ++r) {
      o[r * 4 + 0] = cur[r][0]; o[r * 4 + 1] = cur[r][1]; o[r * 4 + 2] = cur[r][2];
      o[r * 4 + 3] = cur[r][3] -
          (cur[r][0] * Jc[0] + cur[r][1] * Jc[1] + cur[r][2] * Jc[2]);
    }
    o[12] = 0.f; o[13] = 0.f; o[14] = 0.f; o[15] = 1.f;
  };
  emit(0, Jp);
  for (int i = 1; i < NJOINT; ++i) {          // parent(i) == i-1 (pure chain)
    rodrigues(th + 3 * i, R);
    const float Jc[3] = {Jn[3 * i], Jn[3 * i + 1], Jn[3 * i + 2]};
    float si[3], sp[3];
#pragma unroll
    for (int c = 0; c < 3; ++c) { si[c] = scal_of(i, c, e); sp[c] = scal_of(i - 1, c, e); }
    float loc[3][4];
#pragma unroll
    for (int r = 0; r < 3; ++r) {
#pragma unroll
      for (int c = 0; c < 3; ++c) loc[r][c] = R[r][c] * si[c] / sp[r];
      loc[r][3] = Jc[r] - Jp[r];
    }
    float nc[3][4];
#pragma unroll
    for (int r = 0; r < 3; ++r)
#pragma unroll
      for (int c = 0; c < 4; ++c) {
        float acc = cur[r][0] * loc[0][c] + cur[r][1] * loc[1][c] +
                    cur[r][2] * loc[2][c];
        if (c == 3) acc += cur[r][3];
        nc[r][c] = acc;
      }
#pragma unroll
    for (int r = 0; r < 3; ++r)
#pragma unroll
      for (int c = 0; c < 4; ++c) cur[r][c] = nc[r][c];
    emit(i, Jc);
    Jp[0] = Jc[0]; Jp[1] = Jc[1]; Jp[2] = Jc[2];
  }
}

// ---------------------------------------------------------------------------
// K4: skinning. T(16v x 16e) = weights(16v x 33j) @ A[n](33j x 16e) via WMMA,
// K=33: 8 full steps + 1 tail. T tile round-trips LDS; apply in place.
// Clamped vertex index: garbage rows in D are never stored.
// ---------------------------------------------------------------------------
__global__ __launch_bounds__(256) void k4_skin(
    const float* __restrict__ weights, const float* __restrict__ A_ws,
    const float* __restrict__ trans, float* __restrict__ verts) {
  __shared__ float lds[8 * 256];
  const int lane = threadIdx.x & 31;
  const int wave = threadIdx.x >> 5;
  const int n = blockIdx.y;
  const int tile = blockIdx.x * 8 + wave;   // valid tiles: 0..243
  const int hl = lane & 15, hi = lane >> 4;
  const int v = tile * 16 + hl;
  const int vc = (v < NVERT) ? v : (NVERT - 1);
  const float* wrow = weights + vc * NJOINT;
  const float* An = A_ws + (long)n * (NJOINT * 16);
  v8f c = {};
#pragma unroll
  for (int k0 = 0; k0 < 32; k0 += 4) {      // j0+1 <= 31 < 33: no guards
    const int j0 = k0 + 2 * hi;
    v2f a, b;
    a.x = wrow[j0];
    a.y = wrow[j0 + 1];
    b.x = An[j0 * 16 + hl];
    b.y = An[(j0 + 1) * 16 + hl];
    c = wmma_f32x4(a, b, c);
  }
  { // tail k0=32: only hi==0 carries j=32
    const float ta = wrow[32];
    const float tb = An[32 * 16 + hl];
    v2f a, b;
    a.x = hi ? 0.f : ta;  a.y = 0.f;
    b.x = hi ? 0.f : tb;  b.y = 0.f;
    c = wmma_f32x4(a, b, c);
  }
  float* lt = lds + wave * 256;
#pragma unroll
  for (int r = 0; r < 8; ++r) lt[(r + 8 * hi) * 16 + hl] = c[r];
  __syncthreads();
  if (lane < 16) {
    const int vv = tile * 16 + lane;
    if (vv < NVERT) {
      const float* T = lt + lane * 16;       // 4x4 row-major
      const long base = (long)n * V3 + (long)vv * 3;
      const float x = verts[base], y = verts[base + 1], z = verts[base + 2];
      const float tx = trans[n * 3], ty = trans[n * 3 + 1], tz = trans[n * 3 + 2];
      verts[base]     = T[0] * x + T[1] * y + T[2]  * z + T[3]  + tx;
      verts[base + 1] = T[4] * x + T[5] * y + T[6]  * z + T[7]  + ty;
      verts[base + 2] = T[8] * x + T[9] * y + T[10] * z + T[11] + tz;
    }
  }
}

// ---------------------------------------------------------------------------
extern "C" void kernel_launch(void* const* d_in, const int* in_sizes, int n_in,
                              void* d_out, int out_size, void* d_ws,
                              size_t ws_size, hipStream_t stream) {
  (void)in_sizes; (void)n_in; (void)out_size; (void)ws_size;
  const float* beta       = (const float*)d_in[0];
  const float* theta      = (const float*)d_in[1];
  const float* trans      = (const float*)d_in[2];
  const float* blog       = (const float*)d_in[3];
  const float* shapedirs  = (const float*)d_in[4];
  const float* v_template = (const float*)d_in[5];
  const float* Jreg       = (const float*)d_in[6];
  // d_in[7] = posedirs: multiplied by zeros in the reference -> unused.
  const float* weights    = (const float*)d_in[8];

  float* verts  = (float*)d_out;                       // [N, V3] (v_shaped first, then verts in place)
  float* joints = verts + (long)NBATCH * V3;           // [N, 39*3]
  float* J_ws   = (float*)d_ws;                        // [N, 99]
  float* A_ws   = J_ws + (long)NBATCH * NJOINT * 3;    // [N, 528]  (~2.5 MB total ws)

  // 1) v_shaped = v_template + beta @ shapedirs   (WMMA)
  k1_shape<<<dim3(5840), dim3(256), 0, stream>>>(beta, shapedirs, v_template, verts);
  // 2) J = regress(v_shaped)                      (WMMA)
  k_jgemm<<<dim3(72), dim3(256), 0, stream>>>(verts, Jreg, J_ws, 99);
  // 3) kinematic chain -> A matrices
  k3_chain<<<dim3(4), dim3(256), 0, stream>>>(theta, blog, J_ws, A_ws);
  // 4) skinning: T = weights @ A (WMMA), apply in place, add trans
  k4_skin<<<dim3(31, NBATCH), dim3(256), 0, stream>>>(weights, A_ws, trans, verts);
  // 5) joints = regress(verts)                    (WMMA) ++ extra verts
  k_jgemm<<<dim3(72), dim3(256), 0, stream>>>(verts, Jreg, joints, 117);
  k_extras<<<dim3(72), dim3(256), 0, stream>>>(verts, joints);
}